// PPO_85830626443673
// MI455X (gfx1250) — compile-verified
//
#include <hip/hip_runtime.h>
#include <hip/hip_bf16.h>

typedef __attribute__((ext_vector_type(2))) float v2f;
typedef __attribute__((ext_vector_type(8))) float v8f;

#define T_STEPS 65536
#define SDIM 128
#define HID 128
#define NACT 16
#define GAMMA_F 0.98f
#define CLAM 0.931f   /* gamma*lambda */
#define EPSC 0.1f

// ---------------------------------------------------------------------------
// Kernel 1: v(s), v(s') via fp32 WMMA; writes delta[t] = td_target - v(s).
// Block = 128 threads = 4 waves; each wave owns 16 rows for BOTH s and s'.
// W1 staged into LDS pre-permuted into the WMMA B-operand layout.
// ---------------------------------------------------------------------------
__global__ __launch_bounds__(128) void ppo_value_delta_kernel(
    const float* __restrict__ s, const float* __restrict__ sp,
    const float* __restrict__ r, const float* __restrict__ dm,
    const float* __restrict__ W1, const float* __restrict__ b1,
    const float* __restrict__ Wv, const float* __restrict__ bv,
    float* __restrict__ delta)
{
    __shared__ float lds[HID * SDIM];            // 64 KB, B-layout staged W1
    float2* lds2 = (float2*)lds;
    const float2* w12 = (const float2*)W1;

    const int tid = threadIdx.x;
    // Stage W1[j][k] -> B-layout: slot = (ktile*8 + ntile)*32 + lane',
    // lane' = (j&15) + 16*(k-half), float2 = (K even, K odd) of the 4-K tile.
    #pragma unroll 8
    for (int i = 0; i < 64; ++i) {
        int idx    = tid + i * 128;              // float2 index = j*64 + u
        int j      = idx >> 6;
        int u      = idx & 63;
        int ktile  = u >> 1;
        int lane_p = (j & 15) + ((u & 1) << 4);
        int ntile  = j >> 4;
        lds2[(ktile * 8 + ntile) * 32 + lane_p] = w12[idx];
    }
    __syncthreads();

    const int wave = tid >> 5;
    const int lane = tid & 31;
    const int m0   = blockIdx.x * 64 + wave * 16;
    const int half = lane >> 4;
    const int row  = m0 + (lane & 15);

    float vres[2][8];
    const float* Xs[2] = { s, sp };

    for (int p = 0; p < 2; ++p) {
        const float* X = Xs[p] + (size_t)row * SDIM + 2 * half;
        v8f acc[8];
        #pragma unroll
        for (int j = 0; j < 8; ++j) { v8f z = {}; acc[j] = z; }

        #pragma unroll 4
        for (int kt = 0; kt < 32; ++kt) {
            v2f a2 = *(const v2f*)(X + kt * 4);  // A tile: 16x4, lane-major
            #pragma unroll
            for (int j = 0; j < 8; ++j) {
                v2f b2 = *(const v2f*)&lds2[(kt * 8 + j) * 32 + lane];
                acc[j] = __builtin_amdgcn_wmma_f32_16x16x4_f32(
                    false, a2, false, b2, (short)0, acc[j], false, false);
            }
        }

        // h = relu(acc + b1[col]); per-row partial of h * Wv[col]
        float rowsum[8];
        #pragma unroll
        for (int q = 0; q < 8; ++q) rowsum[q] = 0.f;
        #pragma unroll
        for (int j = 0; j < 8; ++j) {
            int   col = j * 16 + (lane & 15);
            float bb  = b1[col];
            float wv  = Wv[col];
            #pragma unroll
            for (int q = 0; q < 8; ++q) {
                float h = fmaxf(acc[j][q] + bb, 0.f);
                rowsum[q] = fmaf(h, wv, rowsum[q]);
            }
        }
        // reduce across the 16 lanes of each half (rows q / q+8)
        #pragma unroll
        for (int q = 0; q < 8; ++q) {
            #pragma unroll
            for (int off = 1; off < 16; off <<= 1)
                rowsum[q] += __shfl_xor(rowsum[q], off, 32);
            vres[p][q] = rowsum[q];
        }
    }

    if ((lane & 15) == 0) {
        float bv0 = bv[0];
        #pragma unroll
        for (int q = 0; q < 8; ++q) {
            int   t   = m0 + q + 8 * half;
            float vs  = vres[0][q] + bv0;
            float vsp = vres[1][q] + bv0;
            float td  = r[t] + GAMMA_F * vsp * dm[t];
            delta[t]  = td - vs;
        }
    }
}

// ---------------------------------------------------------------------------
// Kernel 2: GAE pass 1. 64 waves, 1024 elements each. In-lane Hillis-Steele
// weighted suffix scan per 32-lane segment, sequential carry over 32 segs.
// Also folds the Huber(critic) partial sums (fixed-order butterfly).
// ---------------------------------------------------------------------------
__global__ __launch_bounds__(32) void ppo_gae_pass1(
    float* __restrict__ dl,        // in: delta; out: local scans (in place)
    float* __restrict__ Lsum,      // [64] block-local scan value at block start
    float* __restrict__ huberp)    // [64] Huber partial sums
{
    const int   lane = threadIdx.x;
    const int   base = blockIdx.x * 1024;
    const float C1 = CLAM;
    const float C2 = C1 * C1, C4 = C2 * C2, C8 = C4 * C4, C16 = C8 * C8;
    const float wc = powf(C1, (float)(32 - lane));  // carry weight c^(32-lane)

    float carry = 0.f;
    float hub   = 0.f;
    for (int seg = 31; seg >= 0; --seg) {
        int   idx = base + seg * 32 + lane;
        float d   = dl[idx];
        float ad  = fabsf(d);
        hub += (ad < 1.f) ? 0.5f * d * d : (ad - 0.5f);

        float v = d, u;
        u = __shfl_down(v, 1, 32);  if (lane + 1  < 32) v += C1  * u;
        u = __shfl_down(v, 2, 32);  if (lane + 2  < 32) v += C2  * u;
        u = __shfl_down(v, 4, 32);  if (lane + 4  < 32) v += C4  * u;
        u = __shfl_down(v, 8, 32);  if (lane + 8  < 32) v += C8  * u;
        u = __shfl_down(v, 16, 32); if (lane + 16 < 32) v += C16 * u;
        v += wc * carry;
        dl[idx] = v;
        carry = __shfl(v, 0, 32);
    }
    #pragma unroll
    for (int off = 16; off >= 1; off >>= 1) hub += __shfl_xor(hub, off, 32);
    if (lane == 0) { Lsum[blockIdx.x] = carry; huberp[blockIdx.x] = hub; }
}

// ---------------------------------------------------------------------------
// Kernel 3: sequential combine of 64 block carries (c^1024 ~ 1.6e-32).
// ---------------------------------------------------------------------------
__global__ void ppo_gae_pass2(const float* __restrict__ Lsum,
                              float* __restrict__ Afull)   // [65]
{
    if (threadIdx.x == 0 && blockIdx.x == 0) {
        const float Cblk = powf(CLAM, 1024.0f);
        float a = 0.f;
        Afull[64] = 0.f;
        for (int b = 63; b >= 0; --b) {
            a = Lsum[b] + Cblk * a;
            Afull[b] = a;
        }
    }
}

// ---------------------------------------------------------------------------
// Kernel 4: apply carries -> advantage, PPO clipped surrogate, fixed-order
// block tree reduction to per-block actor partials.
// ---------------------------------------------------------------------------
__global__ __launch_bounds__(256) void ppo_actor_pass(
    const float* __restrict__ locals, const float* __restrict__ Afull,
    const int* __restrict__ a, const float* __restrict__ prob_a,
    const float* __restrict__ pi, float* __restrict__ actorp)
{
    __shared__ float red[256];
    const int   tid   = threadIdx.x;
    const int   b     = blockIdx.x;
    const float carry = Afull[b + 1];

    float acc = 0.f;
    #pragma unroll
    for (int i = 0; i < 4; ++i) {
        int   tl  = tid + i * 256;
        int   t   = b * 1024 + tl;
        float adv = locals[t] + powf(CLAM, (float)(1024 - tl)) * carry;
        float pia   = pi[t * NACT + a[t]];
        float ratio = expf(logf(pia) - logf(prob_a[t]));
        float s1 = ratio * adv;
        float rc = fminf(fmaxf(ratio, 1.f - EPSC), 1.f + EPSC);
        float s2 = rc * adv;
        acc += -fminf(s1, s2);
    }
    red[tid] = acc;
    __syncthreads();
    for (int off = 128; off >= 1; off >>= 1) {
        if (tid < off) red[tid] += red[tid + off];
        __syncthreads();
    }
    if (tid == 0) actorp[b] = red[0];
}

// ---------------------------------------------------------------------------
// Kernel 5: deterministic final combine: mean(-min(surr)) + mean(huber).
// ---------------------------------------------------------------------------
__global__ void ppo_finalize(const float* __restrict__ actorp,
                             const float* __restrict__ huberp,
                             float* __restrict__ out)
{
    if (threadIdx.x == 0 && blockIdx.x == 0) {
        float as = 0.f, hs = 0.f;
        for (int i = 0; i < 64; ++i) { as += actorp[i]; hs += huberp[i]; }
        out[0] = as / (float)T_STEPS + hs / (float)T_STEPS;
    }
}

extern "C" void kernel_launch(void* const* d_in, const int* in_sizes, int n_in,
                              void* d_out, int out_size, void* d_ws, size_t ws_size,
                              hipStream_t stream)
{
    const float* s  = (const float*)d_in[0];
    const int*   a  = (const int*)  d_in[1];
    const float* r  = (const float*)d_in[2];
    const float* sp = (const float*)d_in[3];
    const float* dm = (const float*)d_in[4];
    const float* pa = (const float*)d_in[5];
    const float* pi = (const float*)d_in[6];
    const float* W1 = (const float*)d_in[7];
    const float* b1 = (const float*)d_in[8];
    const float* Wv = (const float*)d_in[9];
    const float* bv = (const float*)d_in[10];

    float* ws     = (float*)d_ws;
    float* delta  = ws;                 // T floats (becomes local scans)
    float* Lsum   = ws + T_STEPS;       // 64
    float* Afull  = Lsum + 64;          // 65
    float* huberp = Afull + 65;         // 64
    float* actorp = huberp + 64;        // 64

    ppo_value_delta_kernel<<<dim3(T_STEPS / 64), dim3(128), 0, stream>>>(
        s, sp, r, dm, W1, b1, Wv, bv, delta);
    ppo_gae_pass1<<<dim3(64), dim3(32), 0, stream>>>(delta, Lsum, huberp);
    ppo_gae_pass2<<<dim3(1), dim3(1), 0, stream>>>(Lsum, Afull);
    ppo_actor_pass<<<dim3(64), dim3(256), 0, stream>>>(delta, Afull, a, pa, pi, actorp);
    ppo_finalize<<<dim3(1), dim3(1), 0, stream>>>(actorp, huberp, (float*)d_out);
}